// BaseCPNN_81432579932692
// MI455X (gfx1250) — compile-verified
//
#include <hip/hip_runtime.h>

// CPNN forward for MI455X (gfx1250, wave32):
//   winners = argmin_h || x_b - kw_h ||   via f32 WMMA GEMM + fused argmin
//   out[b]  = gw[:, winners[b]]           via L2-resident gather
//
// d_in[0] = x  [4096 x 512]  f32
// d_in[1] = kw [16384 x 512] f32
// d_in[2] = gw [1000 x 16384] f32
// d_out   = [4096*1000 floats output][4096 floats winners-as-float]
// d_ws    = [4096 u64 argmin keys][16384 f32 w2][4096 f32 x2]  (112 KB)

#define BATCH 4096
#define IN    512
#define HID   16384
#define OUT   1000

#define BM      16                  // batch rows per block (one 16-row A tile)
#define HSPLIT  2                   // hidden-dim splits (grid.y)
#define WAVES   8                   // 256 threads / wave32
#define CHUNKS  ((HID / HSPLIT) / 16)
#define STEPS   (IN / 8)            // 64 8-wide K steps per chunk
#define HQ      16                  // K steps per pipelined quarter-group

typedef float v2f __attribute__((ext_vector_type(2)));
typedef float v8f __attribute__((ext_vector_type(8)));

__global__ void init_keys_kernel(unsigned long long* __restrict__ keys) {
    int i = blockIdx.x * blockDim.x + threadIdx.x;
    if (i < BATCH) keys[i] = ~0ull;
}

// One pass over kw and x: row-wise sum of squares (f32, matches reference math).
__global__ __launch_bounds__(256)
void rownorm_kernel(const float* __restrict__ kw, const float* __restrict__ x,
                    float* __restrict__ w2, float* __restrict__ x2) {
    const int wv = threadIdx.x >> 5, lane = threadIdx.x & 31;
    const int row = blockIdx.x * 8 + wv;
    const float* src;
    float* dst;
    if (row < HID) {
        src = kw + (size_t)row * IN;
        dst = w2 + row;
    } else {
        int r = row - HID;
        if (r >= BATCH) return;
        src = x + (size_t)r * IN;
        dst = x2 + r;
    }
    const float4* s4 = (const float4*)src;
    float s = 0.0f;
#pragma unroll
    for (int j = 0; j < IN / 128; ++j) {         // 4 x float4 per lane
        float4 v = s4[lane + 32 * j];
        s += v.x * v.x + v.y * v.y + v.z * v.z + v.w * v.w;
    }
#pragma unroll
    for (int off = 16; off >= 1; off >>= 1) s += __shfl_xor(s, off, 32);
    if (lane == 0) *dst = s;
}

__global__ __launch_bounds__(256)
void kohonen_argmin_kernel(const float* __restrict__ x,
                           const float* __restrict__ kw,
                           const float* __restrict__ w2arr,
                           const float* __restrict__ x2arr,
                           unsigned long long* __restrict__ keys) {
    const int tid  = threadIdx.x;
    const int row0 = blockIdx.x * BM;
    const int wv   = tid >> 5;
    const int lane = tid & 31;
    const int h    = lane >> 4;   // lane half (wave32: 2 halves of 16)
    const int l16  = lane & 15;

    // Stationary A operand: x row (row0+l16); half h owns k = 8s+4h..+3 of
    // step s. Held in 256 VGPRs for the whole kernel (chunk-invariant), so the
    // steady-state loop is pure global-B b128 loads + WMMA. One-time 32 KB/wave
    // read; x tile is L1/L2 resident. No LDS, no barrier needed.
    float4 aReg[STEPS];
    {
        const float4* apg = (const float4*)(x + (size_t)(row0 + l16) * IN) + h;
#pragma unroll
        for (int s = 0; s < STEPS; ++s) aReg[s] = apg[s * 2];
    }

    // ||x||^2 for the 8 C-rows this lane half owns (M = v + 8h).
    float x2r[8];
#pragma unroll
    for (int v = 0; v < 8; ++v) x2r[v] = x2arr[row0 + v + 8 * h];

    float bestd[8];
    int   bestn[8];
#pragma unroll
    for (int v = 0; v < 8; ++v) { bestd[v] = 3.4e38f; bestn[v] = 0; }

    const int nsplit0 = blockIdx.y * (HID / HSPLIT);

    // Ping-pong B buffers: 16 steps (16 x b128) per quarter-group. Loads for a
    // group are always issued >= one 32-WMMA crunch before their consumption.
    float4 P[HQ], Q[HQ];

    auto loadQ = [&](int nbase, int q, float4* buf) {
        const float4* wp = (const float4*)(kw + (size_t)(nbase + l16) * IN) + h;
#pragma unroll
        for (int j = 0; j < HQ; ++j) buf[j] = wp[(q * HQ + j) * 2];
    };
    auto crunchQ = [&](int q, const float4* buf, v8f& acc) {
#pragma unroll
        for (int j = 0; j < HQ; ++j) {
            float4 a = aReg[q * HQ + j];
            float4 b = buf[j];
            v2f axy = {a.x, a.y}, azw = {a.z, a.w};
            v2f bxy = {b.x, b.y}, bzw = {b.z, b.w};
            acc = __builtin_amdgcn_wmma_f32_16x16x4_f32(
                false, axy, false, bxy, (short)0, acc, false, false);
            acc = __builtin_amdgcn_wmma_f32_16x16x4_f32(
                false, azw, false, bzw, (short)0, acc, false, false);
        }
    };

    // Each wave owns chunks wv, wv+8, ... (ascending n => argmin keeps lowest
    // index). Software pipeline across quarter-groups and chunk boundaries.
    loadQ(nsplit0 + wv * 16, 0, P);
    for (int ci = wv; ci < CHUNKS; ci += WAVES) {
        const int nbase = nsplit0 + ci * 16;
        const float w2  = w2arr[nbase + l16];   // prefetch epilogue scalar early

        v8f acc = {};
        loadQ(nbase, 1, Q);
        crunchQ(0, P, acc);
        loadQ(nbase, 2, P);
        crunchQ(1, Q, acc);
        loadQ(nbase, 3, Q);
        crunchQ(2, P, acc);
        if (ci + WAVES < CHUNKS) loadQ(nsplit0 + (ci + WAVES) * 16, 0, P);
        crunchQ(3, Q, acc);

        const int n = nbase + l16;
        // C/D layout: element v, lanes 0-15 -> M=v, lanes 16-31 -> M=v+8; N=l16.
#pragma unroll
        for (int v = 0; v < 8; ++v) {
            float d2 = fmaxf(x2r[v] + w2 - 2.0f * acc[v], 0.0f);  // max(d2,0)
            if (d2 < bestd[v]) { bestd[v] = d2; bestn[v] = n; }
        }
    }

    // Reduce across the 16 lanes of each half; pack (d2,index) so u64-min
    // tie-breaks toward the lowest index, like jnp.argmin.
#pragma unroll
    for (int v = 0; v < 8; ++v) {
        unsigned long long key =
            ((unsigned long long)__float_as_uint(bestd[v]) << 32) |
            (unsigned int)bestn[v];
#pragma unroll
        for (int off = 8; off >= 1; off >>= 1) {
            unsigned long long other = __shfl_xor(key, off, 32);
            if (other < key) key = other;
        }
        if (l16 == 0) {
            atomicMin(&keys[row0 + v + 8 * h], key);  // global_atomic_min_u64
        }
    }
}

__global__ __launch_bounds__(256)
void gather_out_kernel(const unsigned long long* __restrict__ keys,
                       const float* __restrict__ gw,
                       float* __restrict__ out) {
    const int b      = blockIdx.x;
    const int winner = (int)(keys[b] & 0xFFFFFFFFull);
    const float* col = gw + winner;                  // gw[:, winner], stride HID
    for (int o = threadIdx.x; o < OUT; o += 256)
        out[(size_t)b * OUT + o] = col[(size_t)o * HID];
    if (threadIdx.x == 0)
        out[(size_t)BATCH * OUT + b] = (float)winner;  // winners tail, as float
}

extern "C" void kernel_launch(void* const* d_in, const int* in_sizes, int n_in,
                              void* d_out, int out_size, void* d_ws, size_t ws_size,
                              hipStream_t stream) {
    const float* x  = (const float*)d_in[0];
    const float* kw = (const float*)d_in[1];
    const float* gw = (const float*)d_in[2];
    float* out = (float*)d_out;

    unsigned long long* keys = (unsigned long long*)d_ws;          // 32 KB
    float* w2 = (float*)((char*)d_ws + BATCH * sizeof(unsigned long long));
    float* x2 = w2 + HID;                                          // +64 KB, +16 KB

    init_keys_kernel<<<(BATCH + 255) / 256, 256, 0, stream>>>(keys);
    rownorm_kernel<<<(HID + BATCH) / 8, 256, 0, stream>>>(kw, x, w2, x2);

    dim3 grid(BATCH / BM, HSPLIT);
    kohonen_argmin_kernel<<<grid, 256, 0, stream>>>(x, kw, w2, x2, keys);

    gather_out_kernel<<<BATCH, 256, 0, stream>>>(keys, gw, out);
}